// kmeans_cluster_assignment_52544629899792
// MI455X (gfx1250) — compile-verified
//
#include <hip/hip_runtime.h>
#include <hip/hip_bf16.h>

// fp32 WMMA path for gfx1250: D(16x16 f32) = A(16x4 f32) x B(4x16 f32) + C
typedef float v2f __attribute__((ext_vector_type(2)));
typedef float v8f __attribute__((ext_vector_type(8)));

#define QPTS 64          // points per problem
#define CD   256         // feature dim
#define CSTR 260         // padded LDS row stride (floats): banks (4q+c)%64 -> conflict-free
#define NITER 64         // sequential EM iterations (== Q)

static __device__ __forceinline__ v8f wmma_f32_k4(v2f a, v2f b, v8f c) {
  // (neg_a, A, neg_b, B, c_mod, C, reuse_a, reuse_b)
  return __builtin_amdgcn_wmma_f32_16x16x4_f32(false, a, false, b, (short)0, c,
                                               false, false);
}

__global__ __launch_bounds__(128, 1)
void kmeans_em_kernel(const float* __restrict__ hs, const float* __restrict__ cin,
                      float* __restrict__ out, int nprob) {
  __shared__ float hsL[QPTS * CSTR];   // 64 x 256 tile, padded rows (65 KB)
  __shared__ float cenL[2 * CSTR];     // current centers
  __shared__ float x2L[QPTS];          // |x|^2, computed once
  __shared__ float xcL[QPTS * 2];      // x . c
  __shared__ float aL[QPTS * 2];       // soft assignment
  __shared__ float c2L[2];
  __shared__ float denL[2];
  __shared__ float zeroL[256];         // zero pad for WMMA lanes with N/M >= 2

  const int tid  = threadIdx.x;
  const int wave = tid >> 5;
  const int lane = tid & 31;
  const int half = lane >> 4;          // which 16-lane half of the wave
  const int ln   = lane & 15;
  const int g    = blockIdx.x;

  const float* hsG  = hs  + (size_t)g * (QPTS * CD);
  const float* cenG = cin + (size_t)g * (2 * CD);
  float* outCen = out + (size_t)g * (2 * CD);
  float* outA   = out + (size_t)nprob * (2 * CD) + (size_t)g * (QPTS * 2);

  // ---- stage hs (64 KB) and centers into LDS once ----
  for (int idx = tid; idx < QPTS * (CD / 4); idx += 128) {
    int r = idx >> 6, c4 = idx & 63;
    ((float4*)hsL)[r * (CSTR / 4) + c4] = ((const float4*)hsG)[(size_t)r * (CD / 4) + c4];
  }
  {
    int r = tid >> 6, c4 = tid & 63;   // 128 threads == 2 rows x 64 float4
    ((float4*)cenL)[r * (CSTR / 4) + c4] = ((const float4*)cenG)[r * (CD / 4) + c4];
  }
  zeroL[tid] = 0.0f;
  zeroL[tid + 128] = 0.0f;
  __syncthreads();

  // ---- x2[q] (constant across iterations) ----
  if (tid < QPTS) {
    const float4* row = (const float4*)(hsL + tid * CSTR);
    float s = 0.f;
    #pragma unroll 8
    for (int i = 0; i < CD / 4; ++i) {
      float4 v = row[i];
      s += v.x * v.x + v.y * v.y + v.z * v.z + v.w * v.w;
    }
    x2L[tid] = s;
  }

  // per-lane fetch bases following ISA WMMA VGPR layouts
  const float* bPtr1 = (ln < 2) ? (cenL + ln * CSTR) : zeroL; // GEMM1 B column (N=center)
  const float* aPtr2 = (ln < 2) ? (aL + ln) : zeroL;          // GEMM2 A row (M=center)
  const int   q0    = wave * 16;                              // GEMM1 q-tile
  const float* hrow = hsL + (q0 + ln) * CSTR + 2 * half;      // A: M=q0+ln, K base 2*half
  const int   c0    = wave * 64;                              // GEMM2: 4 c-tiles per wave
  __syncthreads();

  for (int it = 0; it < NITER; ++it) {
    // ================= Stage A: xc[q][k] = hs . cen^T via WMMA =================
    v8f acc0 = {0.f,0.f,0.f,0.f,0.f,0.f,0.f,0.f};
    v8f acc1 = {0.f,0.f,0.f,0.f,0.f,0.f,0.f,0.f};
    #pragma unroll 4
    for (int cb = 0; cb < CD; cb += 8) {
      v2f a0 = *(const v2f*)(hrow + cb);
      v2f b0 = *(const v2f*)(bPtr1 + cb + 2 * half);
      acc0 = wmma_f32_k4(a0, b0, acc0);
      v2f a1 = *(const v2f*)(hrow + cb + 4);
      v2f b1 = *(const v2f*)(bPtr1 + cb + 4 + 2 * half);
      acc1 = wmma_f32_k4(a1, b1, acc1);
    }
    acc0 += acc1;
    if (ln < 2) {                      // only N=0,1 columns are real centers
      #pragma unroll
      for (int v = 0; v < 8; ++v)      // D layout: VGPR v -> M=v (+8 for upper half)
        xcL[(q0 + v + 8 * half) * 2 + ln] = acc0[v];
    }
    if (tid < 2) {                     // c2[k] = |cen_k|^2 (overlaps with GEMM1 waves)
      const float4* row = (const float4*)(cenL + tid * CSTR);
      float s = 0.f;
      #pragma unroll 8
      for (int i = 0; i < CD / 4; ++i) {
        float4 v = row[i];
        s += v.x * v.x + v.y * v.y + v.z * v.z + v.w * v.w;
      }
      c2L[tid] = s;
    }
    __syncthreads();

    // ================= Stage B: dist -> softmax(-dist) over k=2 =================
    if (tid < QPTS) {
      float xq = x2L[tid];
      float d0 = xq + c2L[0] - 2.f * xcL[tid * 2 + 0];
      float d1 = xq + c2L[1] - 2.f * xcL[tid * 2 + 1];
      float t0 = sqrtf(fmaxf(d0, 1e-12f));
      float t1 = sqrtf(fmaxf(d1, 1e-12f));
      float m  = fminf(t0, t1);
      float e0 = __expf(m - t0);
      float e1 = __expf(m - t1);
      float inv = 1.f / (e0 + e1);
      float a0v = e0 * inv, a1v = e1 * inv;
      aL[tid * 2 + 0] = a0v;
      aL[tid * 2 + 1] = a1v;
      if (it == NITER - 1) {           // assignment from centers BEFORE last update
        outA[tid * 2 + 0] = a0v;
        outA[tid * 2 + 1] = a1v;
      }
    }
    __syncthreads();
    if (tid == 0) {                    // deterministic den reduction (no float atomics)
      float s0 = 0.f, s1 = 0.f;
      const float2* p = (const float2*)aL;
      #pragma unroll 8
      for (int q = 0; q < QPTS; ++q) { float2 v = p[q]; s0 += v.x; s1 += v.y; }
      denL[0] = s0;
      denL[1] = s1;
    }
    __syncthreads();

    // ============ Stage C: num[k][c] = a^T . hs via WMMA; center update ============
    v8f n0 = {0.f,0.f,0.f,0.f,0.f,0.f,0.f,0.f};
    v8f n1 = {0.f,0.f,0.f,0.f,0.f,0.f,0.f,0.f};
    v8f n2 = {0.f,0.f,0.f,0.f,0.f,0.f,0.f,0.f};
    v8f n3 = {0.f,0.f,0.f,0.f,0.f,0.f,0.f,0.f};
    #pragma unroll 2
    for (int qb = 0; qb < QPTS; qb += 4) {
      const int k0 = qb + 2 * half;                 // K base for this half-wave
      v2f av;                                       // A: M=ln (center), K=q
      av.x = aPtr2[2 * k0];
      av.y = aPtr2[2 * k0 + 2];
      const float* r0 = hsL + k0 * CSTR + c0 + ln;  // B: K=q rows, N=c cols
      const float* r1 = r0 + CSTR;
      v2f bv;
      bv.x = r0[0];  bv.y = r1[0];  n0 = wmma_f32_k4(av, bv, n0);
      bv.x = r0[16]; bv.y = r1[16]; n1 = wmma_f32_k4(av, bv, n1);
      bv.x = r0[32]; bv.y = r1[32]; n2 = wmma_f32_k4(av, bv, n2);
      bv.x = r0[48]; bv.y = r1[48]; n3 = wmma_f32_k4(av, bv, n3);
    }
    if (half == 0) {                   // rows M=0,1 live in lanes 0-15, VGPR 0/1
      float dn0 = denL[0] + 1e-7f;
      float dn1 = denL[1] + 1e-7f;
      int c = c0 + ln;
      float w00 = n0[0] / dn0, w01 = n1[0] / dn0, w02 = n2[0] / dn0, w03 = n3[0] / dn0;
      float w10 = n0[1] / dn1, w11 = n1[1] / dn1, w12 = n2[1] / dn1, w13 = n3[1] / dn1;
      cenL[c]           = w00; cenL[c + 16]        = w01;
      cenL[c + 32]      = w02; cenL[c + 48]        = w03;
      cenL[CSTR + c]    = w10; cenL[CSTR + c + 16] = w11;
      cenL[CSTR + c + 32] = w12; cenL[CSTR + c + 48] = w13;
      if (it == NITER - 1) {
        outCen[c]           = w00; outCen[c + 16]      = w01;
        outCen[c + 32]      = w02; outCen[c + 48]      = w03;
        outCen[CD + c]      = w10; outCen[CD + c + 16] = w11;
        outCen[CD + c + 32] = w12; outCen[CD + c + 48] = w13;
      }
    }
    __syncthreads();
  }
}

extern "C" void kernel_launch(void* const* d_in, const int* in_sizes, int n_in,
                              void* d_out, int out_size, void* d_ws, size_t ws_size,
                              hipStream_t stream) {
  const float* hs  = (const float*)d_in[0];   // [L,N,Q,C] fp32
  const float* cen = (const float*)d_in[1];   // [L,N,2,C] fp32
  float* out = (float*)d_out;                 // centers (L*N*2*C) then assignment (L*N*Q*2)
  const int nprob = in_sizes[1] / (2 * CD);   // L*N = 1536
  kmeans_em_kernel<<<dim3(nprob), dim3(128), 0, stream>>>(hs, cen, out, nprob);
}